// GausSplatingHead_4741643895121
// MI455X (gfx1250) — compile-verified
//
#include <hip/hip_runtime.h>
#include <math.h>

#define HH 80
#define WW 80
#define CCH 32
#define LOWPASS_F 0.3f
#define NEAR_F 0.2f
#define SORT_SZ 8192

typedef __attribute__((ext_vector_type(2))) float v2f;
typedef __attribute__((ext_vector_type(8))) float v8f;

// Packed per-(cam,gaussian) record, 32 bytes:
// [0]=u [1]=v [2]=0.5*ia [3]=ib [4]=0.5*ic [5]=opacity*valid [6,7]=pad
// ---------------- Kernel 1: per-(camera,gaussian) projection prep ----------------
__global__ void prep_kernel(const float* __restrict__ pc_xyz,   // (N,3)
                            const float* __restrict__ scales,   // (N,1)
                            const float* __restrict__ rots,     // (N,4)
                            const float* __restrict__ density,  // (N)
                            const float* __restrict__ cam_rot,  // (NC,3,3)
                            const float* __restrict__ cam_trans,// (NC,3)
                            const float* __restrict__ cam_intr, // (NC,4)
                            float* __restrict__ packed,         // (NC*N, 8)
                            float* __restrict__ pkey,           // (NC*N)
                            int N, int NC)
{
    int tid = blockIdx.x * blockDim.x + threadIdx.x;
    if (tid >= N * NC) return;
    int cam = tid / N;
    int n   = tid - cam * N;

    // Gaussian 3D covariance factor Mcov = exp(scale) * R(quat)
    float qw = rots[n*4+0], qx = rots[n*4+1], qy = rots[n*4+2], qz = rots[n*4+3];
    float R3[3][3];
    R3[0][0] = 1.f - 2.f*(qy*qy + qz*qz);
    R3[0][1] = 2.f*(qx*qy - qw*qz);
    R3[0][2] = 2.f*(qx*qz + qw*qy);
    R3[1][0] = 2.f*(qx*qy + qw*qz);
    R3[1][1] = 1.f - 2.f*(qx*qx + qz*qz);
    R3[1][2] = 2.f*(qy*qz - qw*qx);
    R3[2][0] = 2.f*(qx*qz - qw*qy);
    R3[2][1] = 2.f*(qy*qz + qw*qx);
    R3[2][2] = 1.f - 2.f*(qx*qx + qy*qy);
    float s = expf(scales[n]);

    const float* Rc = cam_rot  + cam * 9;
    const float* tt = cam_trans + cam * 3;
    const float* K  = cam_intr + cam * 4;
    float fx = K[0], fy = K[1], cx = K[2], cy = K[3];

    float X = pc_xyz[n*3+0], Y = pc_xyz[n*3+1], Z = pc_xyz[n*3+2];
    float camx = Rc[0]*X + Rc[1]*Y + Rc[2]*Z + tt[0];
    float camy = Rc[3]*X + Rc[4]*Y + Rc[5]*Z + tt[1];
    float camz = Rc[6]*X + Rc[7]*Y + Rc[8]*Z + tt[2];

    float tz    = fmaxf(camz, 1e-6f);
    float valid = (camz > NEAR_F) ? 1.0f : 0.0f;
    float u = fx * camx / tz + cx;
    float v = fy * camy / tz + cy;

    // J (2x3), only nonzero entries
    float itz  = 1.0f / tz;
    float J00  = fx * itz;
    float J02  = -fx * camx * itz * itz;
    float J11  = fy * itz;
    float J12  = -fy * camy * itz * itz;

    // A2 = J @ Rcw   (2x3)
    float A2[2][3];
    #pragma unroll
    for (int k = 0; k < 3; ++k) {
        A2[0][k] = J00 * Rc[0*3+k] + J02 * Rc[2*3+k];
        A2[1][k] = J11 * Rc[1*3+k] + J12 * Rc[2*3+k];
    }
    // Bm = A2 @ (s*R3)   (2x3)
    float Bm[2][3];
    #pragma unroll
    for (int i = 0; i < 2; ++i) {
        #pragma unroll
        for (int k = 0; k < 3; ++k) {
            Bm[i][k] = s * (A2[i][0]*R3[0][k] + A2[i][1]*R3[1][k] + A2[i][2]*R3[2][k]);
        }
    }
    // cov = Bm @ Bm^T  (2x2)
    float a  = Bm[0][0]*Bm[0][0] + Bm[0][1]*Bm[0][1] + Bm[0][2]*Bm[0][2] + LOWPASS_F;
    float b  = Bm[0][0]*Bm[1][0] + Bm[0][1]*Bm[1][1] + Bm[0][2]*Bm[1][2];
    float c  = Bm[1][0]*Bm[1][0] + Bm[1][1]*Bm[1][1] + Bm[1][2]*Bm[1][2] + LOWPASS_F;
    float det = a * c - b * b;
    float idet = 1.0f / det;

    // opacity = softplus(density), fold in validity mask
    float d  = density[n];
    float op = fmaxf(d, 0.0f) + log1pf(expf(-fabsf(d)));
    op *= valid;

    float* pk = packed + (size_t)tid * 8;
    pk[0] = u;
    pk[1] = v;
    pk[2] = 0.5f * (c * idet);    // 0.5*ia folded
    pk[3] = -b * idet;            // ib
    pk[4] = 0.5f * (a * idet);    // 0.5*ic folded
    pk[5] = op;
    pk[6] = 0.0f;
    pk[7] = 0.0f;
    pkey[tid] = tz + (1.0f - valid) * 1.0e6f;
}

// ---------------- Kernel 2: per-camera bitonic argsort (stable via idx tie-break) ----------------
__global__ void sort_kernel(const float* __restrict__ pkey,
                            int* __restrict__ idx_out, int N)
{
    extern __shared__ char smem[];
    float* sk = (float*)smem;
    int*   si = (int*)(smem + SORT_SZ * sizeof(float));
    int cam = blockIdx.x;
    const float* key = pkey + cam * N;

    for (int i = threadIdx.x; i < SORT_SZ; i += blockDim.x) {
        sk[i] = (i < N) ? key[i] : 3.0e38f;
        si[i] = i;
    }
    __syncthreads();

    for (int k = 2; k <= SORT_SZ; k <<= 1) {
        for (int j = k >> 1; j > 0; j >>= 1) {
            for (int i = threadIdx.x; i < SORT_SZ; i += blockDim.x) {
                int ixj = i ^ j;
                if (ixj > i) {
                    bool up = ((i & k) == 0);
                    float ka = sk[i], kb = sk[ixj];
                    int   ja = si[i], jb = si[ixj];
                    bool agtb = (ka > kb) || (ka == kb && ja > jb);
                    if (agtb == up) {
                        sk[i] = kb; sk[ixj] = ka;
                        si[i] = jb; si[ixj] = ja;
                    }
                }
            }
            __syncthreads();
        }
    }
    for (int i = threadIdx.x; i < N; i += blockDim.x)
        idx_out[cam * N + i] = si[i];
}

// ---------------- Kernel 3: gather packed records into depth order ----------------
__global__ void gather_kernel(const float* __restrict__ packed,   // (NC*N,8) unsorted
                              const int* __restrict__ sidx,       // (NC,N)
                              float* __restrict__ packed_s,       // (NC*N,8) sorted
                              int N, int NC)
{
    int tid = blockIdx.x * blockDim.x + threadIdx.x;
    if (tid >= N * NC) return;
    int cam = tid / N;
    int j   = sidx[tid];
    const float4* src = (const float4*)(packed + ((size_t)cam * N + j) * 8);
    float4* dst       = (float4*)(packed_s + (size_t)tid * 8);
    dst[0] = src[0];
    dst[1] = src[1];
}

// ---------------- Kernel 4: render. 1 wave = 32 pixels; WMMA f32 16x16x4 accumulation ----------------
__global__ void __launch_bounds__(128)
render_kernel(const float* __restrict__ feats,     // (N,32) original order
              const float* __restrict__ packed_s,  // (NC*N,8) depth-sorted records
              const int* __restrict__ sidx,        // (NC,N) sorted gaussian indices
              float* __restrict__ out,             // (NC,32,80,80)
              int N)
{
    const int lane = threadIdx.x & 31;
    const int wave = threadIdx.x >> 5;
    const int cam  = blockIdx.y;
    const int tile = blockIdx.x * 4 + wave;   // 32 pixels per wave
    const int pbase = tile * 32;
    const int p = pbase + lane;
    const float fpx = (float)(p % WW);
    const float fpy = (float)(p / WW);

    const float* ps = packed_s + (size_t)cam * N * 8;  // contiguous sorted stream
    const int*   ix = sidx + (size_t)cam * N;

    v8f acc_s0c0 = {0,0,0,0,0,0,0,0};
    v8f acc_s0c1 = {0,0,0,0,0,0,0,0};
    v8f acc_s1c0 = {0,0,0,0,0,0,0,0};
    v8f acc_s1c1 = {0,0,0,0,0,0,0,0};
    float T = 1.0f;
    const bool lo = (lane < 16);
    const int chb = lane & 15;

    for (int g = 0; g < N; g += 4) {
        // idx only needed for the feature gather (A operand)
        int4 iv = *(const int4*)(ix + g);
        float w0, w1, w2, w3;
        // serial front-to-back transmittance scan, one lane per pixel;
        // params are a uniform contiguous stream -> scalar-load clauses
        #define GPROC(gg_, wout) {                                           \
            float4 q0 = *(const float4*)(ps + (size_t)(gg_) * 8);            \
            float4 q1 = *(const float4*)(ps + (size_t)(gg_) * 8 + 4);        \
            float dx = fpx - q0.x, dy = fpy - q0.y;                          \
            float t  = q0.z*dx*dx + q1.x*dy*dy + q0.w*dx*dy;                 \
            float e  = __expf(-fmaxf(t, 0.0f));                              \
            float al = fminf(q1.y * e, 0.99f);                               \
            wout = T * al;  T *= (1.0f - al); }
        GPROC(g+0, w0); GPROC(g+1, w1); GPROC(g+2, w2); GPROC(g+3, w3);
        #undef GPROC

        // Pack weights into B-matrix layout (4x16 f32: VGPR0 = K0|K2, VGPR1 = K1|K3)
        float w2l = __shfl(w2, chb, 32);        // lanes 16-31 read lane-16's w2
        float w3l = __shfl(w3, chb, 32);
        float w0h = __shfl(w0, lane | 16, 32);  // lanes 0-15 read lane+16's w0
        float w1h = __shfl(w1, lane | 16, 32);
        v2f b_s0 = { lo ? w0 : w2l, lo ? w1 : w3l };   // pixels pbase..pbase+15
        v2f b_s1 = { lo ? w0h : w2, lo ? w1h : w3 };   // pixels pbase+16..pbase+31

        // A-matrix (16 channels x 4 gaussians): VGPR0 = K0|K2, VGPR1 = K1|K3
        int glo = lo ? iv.x : iv.z;
        int ghi = lo ? iv.y : iv.w;
        v2f a_c0 = { feats[glo*CCH + chb],      feats[ghi*CCH + chb] };
        v2f a_c1 = { feats[glo*CCH + 16 + chb], feats[ghi*CCH + 16 + chb] };

        acc_s0c0 = __builtin_amdgcn_wmma_f32_16x16x4_f32(false, a_c0, false, b_s0,
                                                         (short)0, acc_s0c0, false, false);
        acc_s0c1 = __builtin_amdgcn_wmma_f32_16x16x4_f32(false, a_c1, false, b_s0,
                                                         (short)0, acc_s0c1, false, false);
        acc_s1c0 = __builtin_amdgcn_wmma_f32_16x16x4_f32(false, a_c0, false, b_s1,
                                                         (short)0, acc_s1c0, false, false);
        acc_s1c1 = __builtin_amdgcn_wmma_f32_16x16x4_f32(false, a_c1, false, b_s1,
                                                         (short)0, acc_s1c1, false, false);

        // Front-to-back saturation: residual contribution bounded by T (<=1e-7)
        if (__all(T < 1.0e-7f)) break;
    }

    // C/D layout: VGPR r = {M=r, lanes 0-15} | {M=r+8, lanes 16-31}, N = lane%16
    const int crow = lo ? 0 : 8;
    const int p0 = pbase + chb;
    float* ob = out + (size_t)cam * CCH * (HH * WW);
    #pragma unroll
    for (int r = 0; r < 8; ++r) {
        ob[(r + crow)      * (HH*WW) + p0]      = acc_s0c0[r];
        ob[(16 + r + crow) * (HH*WW) + p0]      = acc_s0c1[r];
        ob[(r + crow)      * (HH*WW) + p0 + 16] = acc_s1c0[r];
        ob[(16 + r + crow) * (HH*WW) + p0 + 16] = acc_s1c1[r];
    }
}

extern "C" void kernel_launch(void* const* d_in, const int* in_sizes, int n_in,
                              void* d_out, int out_size, void* d_ws, size_t ws_size,
                              hipStream_t stream)
{
    const float* vox      = (const float*)d_in[0]; // (1,N,32)
    const float* density  = (const float*)d_in[1]; // (1,N)
    const float* cam_rot  = (const float*)d_in[2]; // (1,NC,3,3)
    const float* cam_trans= (const float*)d_in[3]; // (1,NC,3)
    const float* cam_intr = (const float*)d_in[4]; // (1,NC,4)
    const float* pc_xyz   = (const float*)d_in[5]; // (N,3)
    const float* scales   = (const float*)d_in[6]; // (N,1)
    const float* rots     = (const float*)d_in[7]; // (N,4)

    const int N  = in_sizes[5] / 3;   // 6144
    const int NC = in_sizes[4] / 4;   // 6
    const int M  = N * NC;

    float* ws       = (float*)d_ws;
    float* packed   = ws;                       // 8*M floats
    float* pkey     = packed + (size_t)8 * M;   // M floats
    int*   sidx     = (int*)(pkey + M);         // M ints
    float* packed_s = (float*)(sidx + M);       // 8*M floats

    prep_kernel<<<(M + 255) / 256, 256, 0, stream>>>(
        pc_xyz, scales, rots, density, cam_rot, cam_trans, cam_intr,
        packed, pkey, N, NC);

    sort_kernel<<<NC, 1024, SORT_SZ * (sizeof(float) + sizeof(int)), stream>>>(
        pkey, sidx, N);

    gather_kernel<<<(M + 255) / 256, 256, 0, stream>>>(
        packed, sidx, packed_s, N, NC);

    dim3 grid((HH * WW / 32) / 4, NC);   // 200 waves/cam, 4 waves/block
    render_kernel<<<grid, 128, 0, stream>>>(
        vox, packed_s, sidx, (float*)d_out, N);
}